// SinkhornClusterLoss_16338055594758
// MI455X (gfx1250) — compile-verified
//
#include <hip/hip_runtime.h>
#include <math.h>

// ---------------------------------------------------------------------------
// Problem constants
// ---------------------------------------------------------------------------
#define B_ROWS  16384      // batch
#define K_CLUST 3000       // clusters
#define K_PAD   3008       // padded to 16 (vector workspaces)
#define K_PAD2  3072       // padded to 128 (GEMM N tiling; nc zero-padded)
#define D_DIM   256        // feature dim
#define INV_EPS 20.0f      // 1/0.05
#define INV_T   10.0f      // 1/0.1
#define LOG_B   9.7040605f // ln(16384)
#define ROWS_PER_CHUNK 128 // rows per column-pass block

typedef __attribute__((ext_vector_type(2))) float v2f;
typedef __attribute__((ext_vector_type(8))) float v8f;

__device__ __forceinline__ float wave_reduce_sum(float x) {
  #pragma unroll
  for (int off = 16; off > 0; off >>= 1)
    x += __shfl_xor(x, off, 32);
  return x;
}

// ---------------------------------------------------------------------------
// 1) Row-wise L2 normalize: out[r,:] = in[r,:] / max(||in[r,:]||, 1e-12)
//    One wave32 per 256-wide row (8 elems/lane). Rows in [nrows, nrows_pad)
//    are written as zeros so the GEMM can read without masking.
// ---------------------------------------------------------------------------
__global__ __launch_bounds__(256) void l2norm_rows(const float* __restrict__ in,
                                                   float* __restrict__ out,
                                                   int nrows, int nrows_pad) {
  const int row  = blockIdx.x * 8 + (threadIdx.x >> 5);
  if (row >= nrows_pad) return;
  const int lane = threadIdx.x & 31;
  float* op = out + (size_t)row * D_DIM;
  if (row >= nrows) {                 // zero-fill padding rows
    #pragma unroll
    for (int j = 0; j < 8; ++j) op[lane + 32 * j] = 0.0f;
    return;
  }
  const float* rp = in + (size_t)row * D_DIM;
  float vals[8];
  float s = 0.f;
  #pragma unroll
  for (int j = 0; j < 8; ++j) {
    float v = rp[lane + 32 * j];
    vals[j] = v;
    s += v * v;
  }
  s = wave_reduce_sum(s);
  const float scale = 1.0f / fmaxf(sqrtf(s), 1e-12f);
  #pragma unroll
  for (int j = 0; j < 8; ++j) op[lane + 32 * j] = vals[j] * scale;
}

// ---------------------------------------------------------------------------
// 2) S = A (row-major [B_ROWS,256]) x C^T (C row-major [K_PAD2,256], rows
//    >= K_CLUST are zero).  Full fp32 via V_WMMA_F32_16X16X4_F32.
//    8 waves / block arranged 2(M) x 4(N); each wave owns a 64x32 strip:
//    4 M-subtiles x 2 N-subtiles = 8 accumulators.  Per K-step: 6 b64 loads
//    feed 8 WMMAs (A frag reused 2x, B frag reused 4x).  The half-wave K
//    offset (2*hh floats) is folded into the base pointers and the K-loop is
//    fully unrolled, so every load is base+imm with no address math -> the
//    scheduler can keep many loads in flight ahead of the WMMAs.
//    A frag (16x4 f32): lanes 0-15 hold (M=lr, K=kk+{0,1}), lanes 16-31 hold
//    (M=lr, K=kk+{2,3}); B symmetric in N (ISA 7.12.2).
// ---------------------------------------------------------------------------
#define WMMA_F32(af, bf, acc) \
  __builtin_amdgcn_wmma_f32_16x16x4_f32(false, (af), false, (bf), (short)0, \
                                        (acc), false, false)

__global__ __launch_bounds__(256) void gemm_nt_wmma(const float* __restrict__ A,
                                                    const float* __restrict__ C,
                                                    float* __restrict__ S) {
  const int lane = threadIdx.x & 31;
  const int wid  = threadIdx.x >> 5;
  const int hh   = lane >> 4;   // half-wave select (K pair)
  const int lr   = lane & 15;

  const int m_wave = blockIdx.x * 128 + (wid & 1) * 64;
  const int n_wave = blockIdx.y * 128 + (wid >> 1) * 32;

  // base pointers with the half-wave K offset pre-folded
  const float* a0 = A + (size_t)(m_wave + lr) * D_DIM + 2 * hh;
  const float* a1 = a0 + (size_t)16 * D_DIM;
  const float* a2 = a0 + (size_t)32 * D_DIM;
  const float* a3 = a0 + (size_t)48 * D_DIM;
  const float* b0 = C + (size_t)(n_wave + lr) * D_DIM + 2 * hh;
  const float* b1 = b0 + (size_t)16 * D_DIM;

  v8f acc00 = {}, acc01 = {}, acc10 = {}, acc11 = {};
  v8f acc20 = {}, acc21 = {}, acc30 = {}, acc31 = {};

  #pragma unroll
  for (int kk = 0; kk < D_DIM; kk += 4) {
    const v2f bf0 = *(const v2f*)(b0 + kk);
    const v2f bf1 = *(const v2f*)(b1 + kk);
    const v2f af0 = *(const v2f*)(a0 + kk);
    const v2f af1 = *(const v2f*)(a1 + kk);
    const v2f af2 = *(const v2f*)(a2 + kk);
    const v2f af3 = *(const v2f*)(a3 + kk);
    acc00 = WMMA_F32(af0, bf0, acc00);
    acc01 = WMMA_F32(af0, bf1, acc01);
    acc10 = WMMA_F32(af1, bf0, acc10);
    acc11 = WMMA_F32(af1, bf1, acc11);
    acc20 = WMMA_F32(af2, bf0, acc20);
    acc21 = WMMA_F32(af2, bf1, acc21);
    acc30 = WMMA_F32(af3, bf0, acc30);
    acc31 = WMMA_F32(af3, bf1, acc31);
  }

  // D layout: VGPR i -> M = base + i + 8*hh, N = lr (ISA 7.12.2)
  const int c0 = n_wave + lr;
  const int c1 = n_wave + 16 + lr;
  #pragma unroll
  for (int i = 0; i < 8; ++i) {
    const int mr = i + 8 * hh;
    if (c0 < K_CLUST) {
      S[(size_t)(m_wave +  0 + mr) * K_CLUST + c0] = acc00[i];
      S[(size_t)(m_wave + 16 + mr) * K_CLUST + c0] = acc10[i];
      S[(size_t)(m_wave + 32 + mr) * K_CLUST + c0] = acc20[i];
      S[(size_t)(m_wave + 48 + mr) * K_CLUST + c0] = acc30[i];
    }
    if (c1 < K_CLUST) {
      S[(size_t)(m_wave +  0 + mr) * K_CLUST + c1] = acc01[i];
      S[(size_t)(m_wave + 16 + mr) * K_CLUST + c1] = acc11[i];
      S[(size_t)(m_wave + 32 + mr) * K_CLUST + c1] = acc21[i];
      S[(size_t)(m_wave + 48 + mr) * K_CLUST + c1] = acc31[i];
    }
  }
}

// ---------------------------------------------------------------------------
// 3a) Column pass: part[chunk][k] = sum_{r in chunk} exp(S[r,k]/eps) * w[r]
//     (w==none for the first, global-normalization pass). Deterministic:
//     per-chunk partials, combined in a fixed-order kernel below.
// ---------------------------------------------------------------------------
__global__ __launch_bounds__(256) void colsum_pass(const float* __restrict__ S,
                                                   const float* __restrict__ w,
                                                   int useW,
                                                   float* __restrict__ part) {
  const int k = blockIdx.x * 256 + threadIdx.x;
  if (k >= K_CLUST) return;
  const int r0 = blockIdx.y * ROWS_PER_CHUNK;
  float acc = 0.f;
  for (int r = r0; r < r0 + ROWS_PER_CHUNK; ++r) {
    float e = __expf(S[(size_t)r * K_CLUST + k] * INV_EPS);
    acc += useW ? e * w[r] : e;
  }
  part[(size_t)blockIdx.y * K_PAD + k] = acc;
}

__global__ __launch_bounds__(256) void combine_cols(const float* __restrict__ part,
                                                    float* __restrict__ colsum) {
  const int k = blockIdx.x * 256 + threadIdx.x;
  if (k >= K_CLUST) return;
  float s = 0.f;
  for (int j = 0; j < B_ROWS / ROWS_PER_CHUNK; ++j)
    s += part[(size_t)j * K_PAD + k];
  colsum[k] = s;
}

// total = sum_k colsum[k]  (only needed before the first u update)
__global__ __launch_bounds__(256) void reduce_total(const float* __restrict__ colsum,
                                                    float* __restrict__ total) {
  __shared__ float sm[8];
  float s = 0.f;
  for (int i = threadIdx.x; i < K_CLUST; i += 256) s += colsum[i];
  s = wave_reduce_sum(s);
  const int lane = threadIdx.x & 31, wid = threadIdx.x >> 5;
  if (lane == 0) sm[wid] = s;
  __syncthreads();
  if (wid == 0) {
    s = (lane < 8) ? sm[lane] : 0.f;
    s = wave_reduce_sum(s);
    if (lane == 0) *total = s;
  }
}

// u[k] = scale / (K * colsum[k]); scale = total on iter 1 (v0 == 1/total)
__global__ __launch_bounds__(256) void finalize_u(const float* __restrict__ colsum,
                                                  const float* __restrict__ totalP,
                                                  int useTotal,
                                                  float* __restrict__ u,
                                                  float* __restrict__ lu) {
  const int k = blockIdx.x * 256 + threadIdx.x;
  if (k >= K_CLUST) return;
  const float scale = useTotal ? *totalP : 1.0f;
  const float uu = scale / ((float)K_CLUST * colsum[k]);
  u[k]  = uu;
  lu[k] = __logf(uu);
}

// ---------------------------------------------------------------------------
// 3b) Row pass: rowsum[b] = sum_k exp(S[b,k]/eps)*u[k]; v[b]=1/(B*rowsum).
//     Final iteration also fuses lse[b] = log sum_k exp(S[b,k]/T) (the
//     log-softmax normalizer of the *other* loss) and log(v).
// ---------------------------------------------------------------------------
__global__ __launch_bounds__(256) void rowpass(const float* __restrict__ S,
                                               const float* __restrict__ u,
                                               float* __restrict__ v,
                                               float* __restrict__ lv,
                                               float* __restrict__ lse,
                                               int withLse) {
  const int row  = blockIdx.x * 8 + (threadIdx.x >> 5);
  const int lane = threadIdx.x & 31;
  const float* rp = S + (size_t)row * K_CLUST;
  float a1 = 0.f, a2 = 0.f;
  for (int k = lane; k < K_CLUST; k += 32) {
    const float s = rp[k];
    a1 += __expf(s * INV_EPS) * u[k];
    if (withLse) a2 += __expf(s * INV_T);
  }
  a1 = wave_reduce_sum(a1);
  if (withLse) a2 = wave_reduce_sum(a2);
  if (lane == 0) {
    const float vv = 1.0f / ((float)B_ROWS * a1);
    v[row] = vv;
    if (withLse) {
      lv[row]  = __logf(vv);
      lse[row] = __logf(a2);
    }
  }
}

// ---------------------------------------------------------------------------
// 4) Loss pass: sum_{b,k} code*(log code - pred + lse[b]) with
//    code = exp(Sc/eps)*v[b]*u[k]*B,  log code = Sc/eps + lv[b]+lu[k]+lnB,
//    pred = Sp/T.  Deterministic block partials.
// ---------------------------------------------------------------------------
__global__ __launch_bounds__(256) void loss_pass(const float* __restrict__ Sc,
                                                 const float* __restrict__ Sp,
                                                 const float* __restrict__ u,
                                                 const float* __restrict__ lu,
                                                 const float* __restrict__ v,
                                                 const float* __restrict__ lv,
                                                 const float* __restrict__ lse,
                                                 float* __restrict__ lossPart) {
  __shared__ float sm[8];
  const int k = blockIdx.x * 256 + threadIdx.x;
  float acc = 0.f;
  if (k < K_CLUST) {
    const float uk = u[k], luk = lu[k];
    const int r0 = blockIdx.y * ROWS_PER_CHUNK;
    for (int r = r0; r < r0 + ROWS_PER_CHUNK; ++r) {
      const float sc = Sc[(size_t)r * K_CLUST + k];
      const float sp = Sp[(size_t)r * K_CLUST + k];
      const float code    = __expf(sc * INV_EPS) * v[r] * uk * (float)B_ROWS;
      const float logcode = sc * INV_EPS + lv[r] + luk + LOG_B;
      acc += code * (logcode - sp * INV_T + lse[r]);
    }
  }
  acc = wave_reduce_sum(acc);
  const int lane = threadIdx.x & 31, wid = threadIdx.x >> 5;
  if (lane == 0) sm[wid] = acc;
  __syncthreads();
  if (threadIdx.x == 0) {
    float s = 0.f;
    #pragma unroll
    for (int i = 0; i < 8; ++i) s += sm[i];
    lossPart[blockIdx.y * gridDim.x + blockIdx.x] = s;
  }
}

__global__ __launch_bounds__(256) void finalize_out(const float* __restrict__ lossPart,
                                                    int n, float* __restrict__ out) {
  __shared__ float sm[8];
  float s = 0.f;
  for (int i = threadIdx.x; i < n; i += 256) s += lossPart[i];
  s = wave_reduce_sum(s);
  const int lane = threadIdx.x & 31, wid = threadIdx.x >> 5;
  if (lane == 0) sm[wid] = s;
  __syncthreads();
  if (threadIdx.x == 0) {
    s = 0.f;
    #pragma unroll
    for (int i = 0; i < 8; ++i) s += sm[i];
    out[0] = s / (2.0f * (float)B_ROWS);   // /B (batchmean) then /2 (average)
  }
}

// ---------------------------------------------------------------------------
// Host launcher
// ---------------------------------------------------------------------------
extern "C" void kernel_launch(void* const* d_in, const int* in_sizes, int n_in,
                              void* d_out, int out_size, void* d_ws, size_t ws_size,
                              hipStream_t stream) {
  (void)in_sizes; (void)n_in; (void)out_size; (void)ws_size;
  const float* q = (const float*)d_in[0];
  const float* k = (const float*)d_in[1];
  const float* c = (const float*)d_in[2];

  float* p = (float*)d_ws;
  float* nq = p; p += (size_t)B_ROWS * D_DIM;
  float* nk = p; p += (size_t)B_ROWS * D_DIM;
  float* nc = p; p += (size_t)K_PAD2 * D_DIM;     // zero-padded rows
  float* Sq = p; p += (size_t)B_ROWS * K_CLUST;
  float* Sk = p; p += (size_t)B_ROWS * K_CLUST;
  const int NCHUNK = B_ROWS / ROWS_PER_CHUNK;       // 128
  float* colpart  = p; p += (size_t)NCHUNK * K_PAD;
  float* colsum   = p; p += K_PAD;
  float* totalP   = p; p += 8;
  float* lossPart = p; p += 4096;
  float* uK  = p; p += K_PAD;  float* luK = p; p += K_PAD;
  float* vK  = p; p += B_ROWS; float* lvK = p; p += B_ROWS; float* lseK = p; p += B_ROWS;
  float* uQ  = p; p += K_PAD;  float* luQ = p; p += K_PAD;
  float* vQ  = p; p += B_ROWS; float* lvQ = p; p += B_ROWS; float* lseQ = p; p += B_ROWS;

  // --- normalize (cluster matrix zero-padded to K_PAD2 rows) ---
  l2norm_rows<<<B_ROWS / 8, 256, 0, stream>>>(q, nq, B_ROWS, B_ROWS);
  l2norm_rows<<<B_ROWS / 8, 256, 0, stream>>>(k, nk, B_ROWS, B_ROWS);
  l2norm_rows<<<K_PAD2 / 8, 256, 0, stream>>>(c, nc, K_CLUST, K_PAD2);

  // --- the two GEMMs (shared by both losses) ---
  dim3 gg(B_ROWS / 128, K_PAD2 / 128);
  gemm_nt_wmma<<<gg, 256, 0, stream>>>(nq, nc, Sq);
  gemm_nt_wmma<<<gg, 256, 0, stream>>>(nk, nc, Sk);

  const dim3 cg((K_CLUST + 255) / 256, NCHUNK);   // column-pass grid
  const int  ublk = (K_CLUST + 255) / 256;

  // --- sinkhorn on one matrix (u/v scaling-vector form) ---
  auto sinkhorn = [&](float* S, float* u, float* lu, float* v, float* lv, float* lse) {
    // iter 1: col sums also give the global normalizer (v0 = 1/total)
    colsum_pass<<<cg, 256, 0, stream>>>(S, nullptr, 0, colpart);
    combine_cols<<<ublk, 256, 0, stream>>>(colpart, colsum);
    reduce_total<<<1, 256, 0, stream>>>(colsum, totalP);
    finalize_u<<<ublk, 256, 0, stream>>>(colsum, totalP, 1, u, lu);
    rowpass<<<B_ROWS / 8, 256, 0, stream>>>(S, u, v, nullptr, nullptr, 0);
    // iter 2
    colsum_pass<<<cg, 256, 0, stream>>>(S, v, 1, colpart);
    combine_cols<<<ublk, 256, 0, stream>>>(colpart, colsum);
    finalize_u<<<ublk, 256, 0, stream>>>(colsum, totalP, 0, u, lu);
    rowpass<<<B_ROWS / 8, 256, 0, stream>>>(S, u, v, nullptr, nullptr, 0);
    // iter 3 (final rowpass fuses log(v) and the other loss' lse)
    colsum_pass<<<cg, 256, 0, stream>>>(S, v, 1, colpart);
    combine_cols<<<ublk, 256, 0, stream>>>(colpart, colsum);
    finalize_u<<<ublk, 256, 0, stream>>>(colsum, totalP, 0, u, lu);
    rowpass<<<B_ROWS / 8, 256, 0, stream>>>(S, u, v, lv, lse, 1);
  };

  sinkhorn(Sk, uK, luK, vK, lvK, lseK);   // code for loss1; lseK feeds loss2
  sinkhorn(Sq, uQ, luQ, vQ, lvQ, lseQ);   // code for loss2; lseQ feeds loss1

  // --- KL losses ---
  const dim3 lg((K_CLUST + 255) / 256, NCHUNK);
  const int  nPartPerLoss = lg.x * lg.y;
  loss_pass<<<lg, 256, 0, stream>>>(Sk, Sq, uK, luK, vK, lvK, lseQ, lossPart);
  loss_pass<<<lg, 256, 0, stream>>>(Sq, Sk, uQ, luQ, vQ, lvQ, lseK,
                                    lossPart + nPartPerLoss);
  finalize_out<<<1, 256, 0, stream>>>(lossPart, 2 * nPartPerLoss, (float*)d_out);
}